// GaussianAdjacencyMatrix_74371653698138
// MI455X (gfx1250) — compile-verified
//
#include <hip/hip_runtime.h>
#include <hip/hip_bf16.h>

// GaussianAdjacencyMatrix on MI455X (gfx1250, wave32).
// B=8, N=2048, D=3. Bandwidth-bound: ~268 MB HBM traffic -> ~11.5us floor.
// Inner products via V_WMMA_F32_16X16X4_F32 (D=3 zero-padded to K=4).
// Un-normalized 16-row block kept in LDS (128 KB of the 320 KB/WGP),
// so masks are read once and the output is written once.

typedef float v2f __attribute__((ext_vector_type(2)));
typedef float v8f __attribute__((ext_vector_type(8)));

#define B_      8
#define N_      2048
#define D_      3
#define TILE    16
#define NTILES  (N_ / TILE)     // 128 column tiles
#define WAVES   8
#define THREADS 256

__launch_bounds__(THREADS)
__global__ void gauss_adj_kernel(const float* __restrict__ coords,
                                 const float* __restrict__ masks,
                                 const float* __restrict__ sigma,
                                 float* __restrict__ out)
{
    __shared__ float A_lds[TILE * N_];   // 128 KB: unnormalized kernel rows
    __shared__ float rowsum[TILE];
    __shared__ float rnorm[TILE];

    const int b      = blockIdx.y;
    const int r0     = blockIdx.x * TILE;
    const int tid    = threadIdx.x;
    const int wave   = tid >> 5;         // wave32
    const int lane   = tid & 31;
    const int laneLo = lane & 15;
    const int hi     = lane >> 4;        // 0: K=0,1 half; 1: K=2,3 half

    const float sig    = sigma[0];
    const float inv_s2 = 1.0f / (sig * sig);

    // Init row sums and row squared-norms.
    if (tid < TILE) {
        rowsum[tid] = 0.0f;
        const float* c = coords + ((size_t)b * N_ + (size_t)(r0 + tid)) * D_;
        rnorm[tid] = c[0] * c[0] + c[1] * c[1] + c[2] * c[2];
    }
    __syncthreads();

    // A fragment (16x4 f32, ISA layout): lanes 0-15 carry {K=0,K=1} of row
    // M=laneLo; lanes 16-31 carry {K=2,K=3}. K=3 is the zero pad (D=3).
    const float* crow = coords + ((size_t)b * N_ + (size_t)(r0 + laneLo)) * D_;
    v2f afrag;
    afrag.x = crow[2 * hi];              // K=0 or K=2
    afrag.y = hi ? 0.0f : crow[1];       // K=1 or K=3 (pad)

    float rn[8];
    #pragma unroll
    for (int v = 0; v < 8; ++v) rn[v] = rnorm[v + 8 * hi];

    float partial[8];
    #pragma unroll
    for (int v = 0; v < 8; ++v) partial[v] = 0.0f;

    // Each wave handles 16 of the 128 column tiles.
    for (int ct = wave; ct < NTILES; ct += WAVES) {
        const int c0  = ct * TILE;
        const int col = c0 + laneLo;

        // B fragment (4x16 f32): lane holds column 'col'; halves carry K pairs.
        const float* ccol = coords + ((size_t)b * N_ + (size_t)col) * D_;
        v2f bfrag;
        bfrag.x = ccol[2 * hi];
        bfrag.y = hi ? 0.0f : ccol[1];
        const float cn = ccol[0] * ccol[0] + ccol[1] * ccol[1] + ccol[2] * ccol[2];

        v8f acc = {};
        // D = A x B + 0 : 16x16 tile of row.col inner products in one op.
        acc = __builtin_amdgcn_wmma_f32_16x16x4_f32(
            /*neg_a=*/false, afrag, /*neg_b=*/false, bfrag,
            /*c_mod=*/(short)0, acc, /*reuse_a=*/false, /*reuse_b=*/false);

        // C/D layout: VGPR v -> M = v + 8*hi, N = laneLo.
        const float* mrow = masks + ((size_t)b * N_ + (size_t)(r0 + 8 * hi)) * N_ + col;
        #pragma unroll
        for (int v = 0; v < 8; ++v) {
            const float d   = rn[v] + cn - 2.0f * acc[v];
            const float m   = mrow[(size_t)v * N_];
            const float val = __expf(-d * inv_s2) * m;
            A_lds[(v + 8 * hi) * N_ + col] = val;
            partial[v] += val;
        }
    }

    // Fold per-lane partial sums into the 16 row sums (ds_add_f32).
    #pragma unroll
    for (int v = 0; v < 8; ++v)
        atomicAdd(&rowsum[v + 8 * hi], partial[v]);

    __syncthreads();

    // Normalize from LDS and store coalesced (float4 = global_store_b128).
    float4*       out4 = (float4*)(out + ((size_t)b * N_ + (size_t)r0) * N_);
    const float4* a4   = (const float4*)A_lds;
    for (int m = 0; m < TILE; ++m) {
        const float inv = 1.0f / (rowsum[m] + 1e-8f);
        for (int i = tid; i < N_ / 4; i += THREADS) {
            float4 vdat = a4[m * (N_ / 4) + i];
            vdat.x *= inv; vdat.y *= inv; vdat.z *= inv; vdat.w *= inv;
            out4[(size_t)m * (N_ / 4) + i] = vdat;
        }
    }
}

extern "C" void kernel_launch(void* const* d_in, const int* in_sizes, int n_in,
                              void* d_out, int out_size, void* d_ws, size_t ws_size,
                              hipStream_t stream) {
    const float* coords = (const float*)d_in[0];  // [8, 2048, 3]  f32
    const float* masks  = (const float*)d_in[1];  // [8, 2048, 2048] f32
    const float* sigma  = (const float*)d_in[2];  // [1] f32
    float*       out    = (float*)d_out;          // [8, 2048, 2048] f32

    dim3 grid(NTILES, B_);   // 128 row-blocks x 8 batches
    gauss_adj_kernel<<<grid, THREADS, 0, stream>>>(coords, masks, sigma, out);
}